// MessagePassing_5978594476045
// MI455X (gfx1250) — compile-verified
//
#include <hip/hip_runtime.h>

#ifndef __has_builtin
#define __has_builtin(x) 0
#endif

#define AS1 __attribute__((address_space(1)))
#define AS3 __attribute__((address_space(3)))

typedef int v2i __attribute__((ext_vector_type(2)));   // 64-bit payload type for the b64 async builtin
typedef AS1 v2i* gv2i_p;                               // global (addrspace 1) pointer to v2i
typedef AS3 v2i* lv2i_p;                               // LDS (addrspace 3) pointer to v2i

struct __align__(16) ll2 { long long x, y; };          // one edge's [dst, src], 16B

#define D        128         // feature dim (floats)
#define ROWB     (D * 4)     // 512 bytes per row
#define TILE     256         // edges staged per block
#define THREADS  256         // 8 waves (wave32)

#if __has_builtin(__builtin_amdgcn_global_load_async_to_lds_b64) && \
    __has_builtin(__builtin_amdgcn_s_wait_asynccnt)
#define HAVE_ASYNC_LDS 1
#else
#define HAVE_ASYNC_LDS 0
#endif

__global__ __launch_bounds__(256) void mp_zero_kernel(float4* __restrict__ out4, int n4) {
    int i = blockIdx.x * blockDim.x + threadIdx.x;
    if (i < n4) out4[i] = float4{0.0f, 0.0f, 0.0f, 0.0f};
}

__global__ __launch_bounds__(THREADS)
void mp_scatter_kernel(const float* __restrict__ x,
                       const long long* __restrict__ ei,   // [2, E] int64
                       float* __restrict__ out,
                       int E)
{
    // LDS tile: interleaved [dst, src] int64 pairs, 16B per edge -> 4 KB
    __shared__ __align__(16) long long pair[2 * TILE];

    const int t      = threadIdx.x;
    const int tile0  = blockIdx.x * TILE;
    const int e_glob = tile0 + t;

    const long long* dstp = ei;        // edge_index[0] : destinations
    const long long* srcp = ei + E;    // edge_index[1] : sources

    if (e_glob < E) {
#if HAVE_ASYNC_LDS
        // Async copy engine: per-lane gather of this tile's indices into LDS.
        __builtin_amdgcn_global_load_async_to_lds_b64(
            (gv2i_p)(dstp + e_glob), (lv2i_p)&pair[2 * t], 0, 0);
        __builtin_amdgcn_global_load_async_to_lds_b64(
            (gv2i_p)(srcp + e_glob), (lv2i_p)&pair[2 * t + 1], 0, 0);
#else
        pair[2 * t]     = dstp[e_glob];
        pair[2 * t + 1] = srcp[e_glob];
#endif
    }
#if HAVE_ASYNC_LDS
    __builtin_amdgcn_s_wait_asynccnt(0);
#endif
    __syncthreads();

    const int wave   = __builtin_amdgcn_readfirstlane(t >> 5);  // 0..7, SGPR
    const int lane   = t & 31;
    const int ebase  = wave * 32;                               // wave's 32 edges in tile
    const int nvalid = (E - tile0 < TILE) ? (E - tile0) : TILE;
    const unsigned voff = (unsigned)(lane * 16);                // lane's 16B chunk within a row

    #pragma unroll 4
    for (int i = 0; i < 32; ++i) {
        const int eslot = ebase + i;
        if (eslot >= nvalid) break;                 // wave-uniform branch

        // One ds_load_b128: this edge's [dst, src] (broadcast, uniform per wave).
        const ll2 p2 = ((const ll2*)pair)[eslot];
        const int d = (int)p2.x;
        const int s = (int)p2.y;

        // Pull a future edge's source row toward the WGP cache (low word of src idx).
        const int ahead = eslot + 8;
        if (ahead < nvalid) {
            const int sf = ((const int*)pair)[4 * ahead + 2];
            __builtin_prefetch((const char*)x + (size_t)sf * ROWB + voff, 0, 1);
        }

        // Force uniform row bases into SGPRs -> saddr + per-lane voffset addressing.
        const int soff = __builtin_amdgcn_readfirstlane(s * ROWB);   // <= 25.6MB, fits i32
        const int doff = __builtin_amdgcn_readfirstlane(d * ROWB);

        // One wave covers the whole 512B row: lane -> float4 chunk.
        const float4 v = *(const float4*)((const char*)x + soff + voff);

        const float* ob = (const float*)((const char*)out + doff);

        // Guaranteed hardware no-return f32 atomics, device scope (L2 atomic units).
        asm volatile("global_atomic_add_f32 %0, %1, %2 offset:0  scope:SCOPE_DEV"
                     :: "v"(voff), "v"(v.x), "s"(ob));
        asm volatile("global_atomic_add_f32 %0, %1, %2 offset:4  scope:SCOPE_DEV"
                     :: "v"(voff), "v"(v.y), "s"(ob));
        asm volatile("global_atomic_add_f32 %0, %1, %2 offset:8  scope:SCOPE_DEV"
                     :: "v"(voff), "v"(v.z), "s"(ob));
        asm volatile("global_atomic_add_f32 %0, %1, %2 offset:12 scope:SCOPE_DEV"
                     :: "v"(voff), "v"(v.w), "s"(ob));
    }
}

extern "C" void kernel_launch(void* const* d_in, const int* in_sizes, int n_in,
                              void* d_out, int out_size, void* d_ws, size_t ws_size,
                              hipStream_t stream) {
    (void)n_in; (void)d_ws; (void)ws_size;

    const float*     x  = (const float*)d_in[0];
    const long long* ei = (const long long*)d_in[1];   // [2, E] int64
    float*           out = (float*)d_out;

    const int E = in_sizes[1] / 2;        // 800,000

    // 1) zero the output (harness poisons d_out before timing)
    const int n4 = out_size / 4;          // 6.4M floats -> 1.6M float4
    mp_zero_kernel<<<(n4 + 255) / 256, 256, 0, stream>>>((float4*)d_out, n4);

    // 2) gather + atomic scatter-add
    const int nblocks = (E + TILE - 1) / TILE;   // 3125
    mp_scatter_kernel<<<nblocks, THREADS, 0, stream>>>(x, ei, out, E);
}